// FRC3DLoss_49151605736008
// MI455X (gfx1250) — compile-verified
//
#include <hip/hip_runtime.h>
#include <cmath>

// FRC3D loss for [B=4, 1, D=64, H=256, W=256] f32 volumes on gfx1250.
// 3 FFT passes (W, H, D); D pass fused into the radial-shell reduction.
// 256-pt FFT = radix-16 via V_WMMA_F32_16X16X4_F32 (full fp32).

#define D_DIM 64
#define H_DIM 256
#define W_DIM 256
#define PLANE (H_DIM * W_DIM)      // 65536
#define VOL   (D_DIM * PLANE)      // 4194304
#define BATCH 4
#define RSH   32                   // R = min(D,H,W)/2
#define NBIN  33                   // RSH shells + dump bin
#define TWO_PI 6.28318530717958647692f
#define SSTR  257                  // padded stick stride (LDS bank-conflict free)

typedef __attribute__((ext_vector_type(2))) float v2f;
typedef __attribute__((ext_vector_type(8))) float v8f;

#define AS1 __attribute__((address_space(1)))
#define AS3 __attribute__((address_space(3)))

#if defined(__has_builtin)
#if __has_builtin(__builtin_amdgcn_global_load_async_to_lds_b32) && \
    __has_builtin(__builtin_amdgcn_s_wait_asynccnt)
#define USE_ASYNC_LDS 1
#endif
#endif
#ifndef USE_ASYNC_LDS
#define USE_ASYNC_LDS 0
#endif

static __device__ __forceinline__ v8f wmma4(v2f a, v2f b, v8f c) {
  // D(16x16,f32) = A(16x4,f32) x B(4x16,f32) + C  -- full fp32 WMMA
  return __builtin_amdgcn_wmma_f32_16x16x4_f32(false, a, false, b,
                                               (short)0, c, false, false);
}

// In-place 256-point complex DFT over LDS arrays sr/si (natural order in,
// natural order out). One wave (32 lanes) of work; uses block barriers, so
// every wave in the block must call it the same number of times. Radix-16:
//   M[n1][n2]=x[16*n1+n2];  Y = F16*M;  Y'= Y .* W256^(k1*n2);  Z = Y'*F16;
//   X[k1+16*k2] = Z[k1][k2]
static __device__ __forceinline__ void dft256_wave(float* sr, float* si, int lane) {
  const int half = lane >> 4;
  const int lr   = lane & 15;

  // ---- stage 1: Y = F16 * M ----
  v8f rr = {}, ri = {}, ir = {}, ii = {};
  for (int kc = 0; kc < 4; ++kc) {
    v2f ar, ai, br, bi;
#pragma unroll
    for (int j = 0; j < 2; ++j) {
      const int kk = 4 * kc + 2 * half + j;          // n1
      float sv, cv;
      __sincosf(-TWO_PI * (float)((lr * kk) & 15) / 16.0f, &sv, &cv);
      ar[j] = cv; ai[j] = sv;                        // F16[lr][kk]
      br[j] = sr[16 * kk + lr];                      // M[n1][n2=lr]
      bi[j] = si[16 * kk + lr];
    }
    rr = wmma4(ar, br, rr);
    ii = wmma4(ai, bi, ii);
    ri = wmma4(ar, bi, ri);
    ir = wmma4(ai, br, ir);
  }
  __syncthreads();
  // twiddle + write Y' back to LDS (layout y[k1*16+n2])
#pragma unroll
  for (int v = 0; v < 8; ++v) {
    const int k1 = v + 8 * half;
    const int n2 = lr;
    const float yr = rr[v] - ii[v];
    const float yi = ri[v] + ir[v];
    float sv, cv;
    __sincosf(-TWO_PI * (float)((k1 * n2) & 255) / 256.0f, &sv, &cv);
    sr[16 * k1 + n2] = yr * cv - yi * sv;
    si[16 * k1 + n2] = yr * sv + yi * cv;
  }
  __syncthreads();

  // ---- stage 2: Z = Y' * F16 ----
  v8f rr2 = {}, ri2 = {}, ir2 = {}, ii2 = {};
  for (int kc = 0; kc < 4; ++kc) {
    v2f ar, ai, br, bi;
#pragma unroll
    for (int j = 0; j < 2; ++j) {
      const int kk = 4 * kc + 2 * half + j;          // n2
      ar[j] = sr[16 * lr + kk];                      // Y'[lr][kk]
      ai[j] = si[16 * lr + kk];
      float sv, cv;
      __sincosf(-TWO_PI * (float)((kk * lr) & 15) / 16.0f, &sv, &cv);
      br[j] = cv; bi[j] = sv;                        // F16[kk][lr]
    }
    rr2 = wmma4(ar, br, rr2);
    ii2 = wmma4(ai, bi, ii2);
    ri2 = wmma4(ar, bi, ri2);
    ir2 = wmma4(ai, br, ir2);
  }
  __syncthreads();
#pragma unroll
  for (int v = 0; v < 8; ++v) {
    const int k1 = v + 8 * half;
    const int k2 = lr;
    const int kout = k1 + 16 * k2;                   // natural output order
    sr[kout] = rr2[v] - ii2[v];
    si[kout] = ri2[v] + ir2[v];
  }
  __syncthreads();
}

// Pass 1: FFT along W (contiguous). 256 threads, 8 rows/block, one wave/row.
__global__ __launch_bounds__(256)
void fft256_rows_real(const float* __restrict__ src, float2* __restrict__ dst) {
  __shared__ float sr[8 * 256], si[8 * 256];
  const int tid  = threadIdx.x;
  const int lane = tid & 31;
  const int wave = tid >> 5;
  const size_t base = (size_t)blockIdx.x * (8 * 256);
#pragma unroll
  for (int t = 0; t < 8; ++t) {
    const int idx = tid + 256 * t;                   // 0..2047, coalesced
#if USE_ASYNC_LDS
    __builtin_amdgcn_global_load_async_to_lds_b32(
        (AS1 int*)(src + base + idx),
        (AS3 int*)&sr[idx], 0, 0);
#else
    sr[idx] = src[base + idx];
#endif
    si[idx] = 0.0f;
  }
#if USE_ASYNC_LDS
  __builtin_amdgcn_s_wait_asynccnt(0);
#endif
  __syncthreads();
  dft256_wave(sr + wave * 256, si + wave * 256, lane);
#pragma unroll
  for (int t = 0; t < 8; ++t) {
    const int idx = tid + 256 * t;
    dst[base + idx] = make_float2(sr[idx], si[idx]);
  }
}

// Pass 2: FFT along H (stride W), in place. 256 threads per block process a
// 16(w) x 256(h) tile: coalesced 128B row segments, transpose via LDS with
// padded stick stride; each of the 8 waves runs the FFT on 2 sticks.
__global__ __launch_bounds__(256)
void fft256_h(float2* __restrict__ data) {
  __shared__ float sr[16 * SSTR], si[16 * SSTR];
  const int tid  = threadIdx.x;
  const int lane = tid & 31;
  const int wave = tid >> 5;
  const int d  = blockIdx.x >> 4;           // / (W/16)
  const int w0 = (blockIdx.x & 15) << 4;
  const size_t base = (size_t)d * PLANE + w0;
#pragma unroll 4
  for (int it = 0; it < 16; ++it) {
    const int idx = tid + 256 * it;         // 0..4095
    const int h = idx >> 4;
    const int s = idx & 15;
    const float2 v = data[base + (size_t)h * W_DIM + s];
    sr[s * SSTR + h] = v.x;
    si[s * SSTR + h] = v.y;
  }
  __syncthreads();
#pragma unroll
  for (int p = 0; p < 2; ++p) {             // uniform: barriers stay aligned
    const int stick = wave * 2 + p;
    dft256_wave(sr + stick * SSTR, si + stick * SSTR, lane);
  }
#pragma unroll 4
  for (int it = 0; it < 16; ++it) {
    const int idx = tid + 256 * it;
    const int h = idx >> 4;
    const int s = idx & 15;
    data[base + (size_t)h * W_DIM + s] = make_float2(sr[s * SSTR + h],
                                                     si[s * SSTR + h]);
  }
}

// Pass 3: FFT along D (length 64) as a direct F64 matmul batched over 16
// consecutive w-sticks per wave, fused with the radial-shell reduction.
// gbins layout: [3][BATCH][NBIN] f32.
__global__ __launch_bounds__(32)
void fft64_reduce(const float2* __restrict__ F1, const float2* __restrict__ F2,
                  float* __restrict__ gbins, int b) {
  __shared__ float b1r[D_DIM * 16], b1i[D_DIM * 16];
  __shared__ float b2r[D_DIM * 16], b2i[D_DIM * 16];
  __shared__ float lbins[3 * NBIN];

  const int lane = threadIdx.x;
  const int half = lane >> 4;
  const int lr   = lane & 15;
  const int h  = blockIdx.x >> 4;           // / (W/16)
  const int w0 = (blockIdx.x & 15) << 4;

  for (int i = lane; i < 3 * NBIN; i += 32) lbins[i] = 0.0f;

  // stage 64x16 complex tiles of both spectra into LDS
#pragma unroll 4
  for (int t = 0; t < 32; ++t) {
    const int idx = lane + 32 * t;          // 0..1023
    const int n = idx >> 4;                 // d
    const int s = idx & 15;                 // stick (w offset)
    const size_t addr = (size_t)n * PLANE + (size_t)h * W_DIM + w0 + s;
    __builtin_prefetch(&F2[addr], 0, 3);
    const float2 a = F1[addr];
    const float2 c = F2[addr];
    b1r[idx] = a.x; b1i[idx] = a.y;
    b2r[idx] = c.x; b2i[idx] = c.y;
  }
  __syncthreads();

  const int fh = (h  < H_DIM / 2) ? h  : h  - H_DIM;
  const int fw0 = w0 + lr;
  const int fw = (fw0 < W_DIM / 2) ? fw0 : fw0 - W_DIM;
  const int rhw2 = fh * fh + fw * fw;

  for (int tk = 0; tk < 4; ++tk) {          // output k tiles of 16 rows
    v8f f1re, f1im, f2re, f2im;
    for (int which = 0; which < 2; ++which) {
      v8f rr = {}, ri = {}, ir = {}, ii = {};
      const float* brs = which ? b2r : b1r;
      const float* bis = which ? b2i : b1i;
      for (int kc = 0; kc < 16; ++kc) {     // K = 64 in chunks of 4
        v2f ar, ai, br, bi;
#pragma unroll
        for (int j = 0; j < 2; ++j) {
          const int kk = 4 * kc + 2 * half + j;       // n (d index)
          const int m  = 16 * tk + lr;                // output row k
          float sv, cv;
          __sincosf(-TWO_PI * (float)((m * kk) & 63) / 64.0f, &sv, &cv);
          ar[j] = cv; ai[j] = sv;                     // F64[m][kk]
          br[j] = brs[kk * 16 + lr];
          bi[j] = bis[kk * 16 + lr];
        }
        rr = wmma4(ar, br, rr);
        ii = wmma4(ai, bi, ii);
        ri = wmma4(ar, bi, ri);
        ir = wmma4(ai, br, ir);
      }
      if (which == 0) { f1re = rr - ii; f1im = ri + ir; }
      else            { f2re = rr - ii; f2im = ri + ir; }
    }
    // fused shell accumulation for this 16x16 tile
#pragma unroll
    for (int v = 0; v < 8; ++v) {
      const int kd = 16 * tk + v + 8 * half;
      const int fd = (kd < D_DIM / 2) ? kd : kd - D_DIM;
      const int r2 = fd * fd + rhw2;
      int rs = (int)ceilf(sqrtf((float)r2));
      while (rs * rs < r2) ++rs;
      while (rs >= 1 && (rs - 1) * (rs - 1) >= r2) --rs;
      const int s1 = rs - 1;                 // shell, valid 1..RSH
      const int bin = (s1 >= 1 && s1 <= RSH) ? (s1 - 1) : RSH;
      const float a1r = f1re[v], a1i = f1im[v];
      const float a2r = f2re[v], a2i = f2im[v];
      atomicAdd(&lbins[0 * NBIN + bin], a1r * a2r + a1i * a2i);  // Re(f1*conj(f2))
      atomicAdd(&lbins[1 * NBIN + bin], a1r * a1r + a1i * a1i);  // |f1|^2
      atomicAdd(&lbins[2 * NBIN + bin], a2r * a2r + a2i * a2i);  // |f2|^2
    }
  }
  __syncthreads();
  for (int i = lane; i < 3 * NBIN; i += 32) {
    const int arr = i / NBIN, bin = i % NBIN;
    atomicAdd(&gbins[(arr * BATCH + b) * NBIN + bin], lbins[i]);
  }
}

__global__ void zero_bins(float* __restrict__ gbins) {
  for (int i = threadIdx.x; i < 3 * BATCH * NBIN; i += blockDim.x) gbins[i] = 0.0f;
}

__global__ __launch_bounds__(32)
void finalize(const float* __restrict__ gbins, float* __restrict__ out) {
  __shared__ float ys[RSH];
  const int r = threadIdx.x;                // 0..31 == shells
  float acc = 0.0f;
  for (int bb = 0; bb < BATCH; ++bb) {
    const float c1 = gbins[(0 * BATCH + bb) * NBIN + r];
    const float c2 = gbins[(1 * BATCH + bb) * NBIN + r];
    const float c3 = gbins[(2 * BATCH + bb) * NBIN + r];
    float frc = c1 / sqrtf(c2 * c3);
    if (!isfinite(frc)) frc = 0.0f;
    acc += frc;
  }
  ys[r] = acc * (1.0f / BATCH);
  __syncthreads();
  if (r == 0) {
    float s = 0.0f;
    for (int i = 0; i < RSH - 1; ++i) s += ys[i] + ys[i + 1];
    s *= 0.5f / (float)RSH;                 // dt = 1/R, trapezoid
    out[0] = 1.0f - s;
  }
}

extern "C" void kernel_launch(void* const* d_in, const int* in_sizes, int n_in,
                              void* d_out, int out_size, void* d_ws, size_t ws_size,
                              hipStream_t stream) {
  (void)in_sizes; (void)n_in; (void)out_size; (void)ws_size;
  const float* pred = (const float*)d_in[0];
  const float* targ = (const float*)d_in[1];
  float* out = (float*)d_out;

  // workspace: two complex volumes (per-batch reuse) + bins
  float2* F1 = (float2*)d_ws;
  float2* F2 = F1 + (size_t)VOL;
  float* gbins = (float*)(F2 + (size_t)VOL);

  zero_bins<<<1, 256, 0, stream>>>(gbins);

  for (int b = 0; b < BATCH; ++b) {
    const size_t voff = (size_t)b * VOL;
    // W-axis FFT (real input): 8 rows per 256-thread block
    fft256_rows_real<<<D_DIM * H_DIM / 8, 256, 0, stream>>>(pred + voff, F1);
    fft256_rows_real<<<D_DIM * H_DIM / 8, 256, 0, stream>>>(targ + voff, F2);
    // H-axis FFT (in place): 16-stick tiles, coalesced via LDS transpose
    fft256_h<<<D_DIM * (W_DIM / 16), 256, 0, stream>>>(F1);
    fft256_h<<<D_DIM * (W_DIM / 16), 256, 0, stream>>>(F2);
    // D-axis FFT + fused shell reduction
    fft64_reduce<<<H_DIM * (W_DIM / 16), 32, 0, stream>>>(F1, F2, gbins, b);
  }
  finalize<<<1, 32, 0, stream>>>(gbins, out);
}